// GAT_59751585022056
// MI455X (gfx1250) — compile-verified
//
#include <hip/hip_runtime.h>
#include <hip/hip_bf16.h>
#include <math.h>

#define N_NODES 50000
#define N_EDGES 800000
#define E_TOT   (N_EDGES + N_NODES)   // self-loops appended
#define IN_CH   128
#define HID     32
#define HEADS   4
#define H1DIM   (HEADS * HID)         // 128
#define C2      16
#define OUT_CH  64
#define N_GRAPHS 64

typedef float v2f __attribute__((ext_vector_type(2)));
typedef float v8f __attribute__((ext_vector_type(8)));

// ---- workspace layout (in floats) ----
#define OFF_H1   ((size_t)0)                       // 50000*128
#define OFF_AGG1 (OFF_H1   + (size_t)N_NODES*H1DIM)
#define OFF_AS1  (OFF_AGG1 + (size_t)N_NODES*H1DIM)
#define OFF_AD1  (OFF_AS1  + (size_t)N_NODES*HEADS)
#define OFF_M1   (OFF_AD1  + (size_t)N_NODES*HEADS)
#define OFF_S1   (OFF_M1   + (size_t)N_NODES*HEADS)
#define OFF_G2   (OFF_S1   + (size_t)N_NODES*HEADS)
#define OFF_AGG2 (OFF_G2   + (size_t)N_NODES*C2)
#define OFF_AS2  (OFF_AGG2 + (size_t)N_NODES*C2)
#define OFF_AD2  (OFF_AS2  + (size_t)N_NODES)
#define OFF_M2   (OFF_AD2  + (size_t)N_NODES)
#define OFF_S2   (OFF_M2   + (size_t)N_NODES)
#define OFF_POOL (OFF_S2   + (size_t)N_NODES)      // 64*16
#define OFF_CNT  (OFF_POOL + (size_t)N_GRAPHS*C2)  // 64

__device__ __forceinline__ float lrelu(float x) { return x > 0.f ? x : 0.2f * x; }

__device__ __forceinline__ void edge_sd(const int* __restrict__ ei, int e, int& s, int& d) {
    if (e < N_EDGES) { s = ei[e]; d = ei[N_EDGES + e]; }
    else             { s = e - N_EDGES; d = s; }
}

__device__ float atomicMaxF(float* addr, float val) {
    int old = __float_as_int(*addr);
    while (__int_as_float(old) < val) {
        int assumed = old;
        old = atomicCAS((int*)addr, assumed, __float_as_int(val));
        if (old == assumed) break;
    }
    return __int_as_float(old);
}

// ---- init: zero accumulators, -inf for segment maxes ----
__global__ void init_ws(float* __restrict__ ws) {
    size_t tid = (size_t)blockIdx.x * blockDim.x + threadIdx.x;
    size_t stride = (size_t)gridDim.x * blockDim.x;
    for (size_t i = tid; i < (size_t)N_NODES * H1DIM; i += stride) ws[OFF_AGG1 + i] = 0.f;
    for (size_t i = tid; i < (size_t)N_NODES * HEADS; i += stride) {
        ws[OFF_M1 + i] = -INFINITY; ws[OFF_S1 + i] = 0.f;
    }
    for (size_t i = tid; i < (size_t)N_NODES * C2; i += stride) ws[OFF_AGG2 + i] = 0.f;
    for (size_t i = tid; i < (size_t)N_NODES; i += stride) {
        ws[OFF_M2 + i] = -INFINITY; ws[OFF_S2 + i] = 0.f;
    }
    for (size_t i = tid; i < (size_t)N_GRAPHS * C2; i += stride) ws[OFF_POOL + i] = 0.f;
    for (size_t i = tid; i < (size_t)N_GRAPHS; i += stride) ws[OFF_CNT + i] = 0.f;
}

// ---- fp32 WMMA GEMM, B staged in LDS via async copy ----
// One wave computes a 16 x (TN*16) row-block: each A fragment feeds TN WMMAs.
// B (KK x TN*16) is loaded once per workgroup with GLOBAL_LOAD_ASYNC_TO_LDS_B128
// (ASYNCcnt-tracked, bypasses VGPRs), then read as ds_load.
// 16x16x4 f32 layouts: A 16x4 (lane%16 = M row; VGPR j holds K=(lane/16)*2+j),
// B 4x16 mirrored, C/D: VGPR j -> M = j + (lane/16)*8, N = lane%16.
template <int TN, int KK>
__global__ __launch_bounds__(256) void wmma_gemm_lds(const float* __restrict__ A,
                                                     const float* __restrict__ Bg,
                                                     float* __restrict__ C, int Mtiles) {
    constexpr int NN = TN * 16;
    __shared__ float Bs[KK * NN];

    // cooperative async B -> LDS (all 256 threads, exact multiple => EXEC all ones)
    constexpr int VECS = KK * NN / 4;  // float4 elements
    for (int i = threadIdx.x; i < VECS; i += 256) {
        unsigned int laddr = (unsigned int)(size_t)(&Bs[i * 4]);      // flat addr[31:0] == LDS offset
        unsigned long long gaddr = (unsigned long long)(size_t)(Bg + i * 4);
        asm volatile("global_load_async_to_lds_b128 %0, %1, off"
                     :: "v"(laddr), "v"(gaddr) : "memory");
    }
    asm volatile("s_wait_asynccnt 0x0" ::: "memory");
    __syncthreads();

    int wave = threadIdx.x >> 5;
    int lane = threadIdx.x & 31;
    int mTile = blockIdx.x * 8 + wave;
    if (mTile < Mtiles) {              // wave-uniform; EXEC stays all-ones inside
        int nlane = lane & 15;
        int kh = (lane >> 4) << 1;     // 0 or 2
        const float* Ar = A + (size_t)((mTile << 4) + nlane) * KK;
        v8f acc[TN];
#pragma unroll
        for (int t = 0; t < TN; ++t) acc[t] = (v8f){0.f,0.f,0.f,0.f,0.f,0.f,0.f,0.f};
#pragma unroll 2
        for (int k = 0; k < KK; k += 4) {
            v2f a = *(const v2f*)(Ar + k + kh);   // one b64 load feeds TN WMMAs
#pragma unroll
            for (int t = 0; t < TN; ++t) {
                v2f b;
                b.x = Bs[(k + kh) * NN + t * 16 + nlane];
                b.y = Bs[(k + kh + 1) * NN + t * 16 + nlane];
                acc[t] = __builtin_amdgcn_wmma_f32_16x16x4_f32(
                    false, a, false, b, (short)0, acc[t], false, false);
            }
        }
        int mBase = (mTile << 4) + ((lane >> 4) << 3);
#pragma unroll
        for (int t = 0; t < TN; ++t)
#pragma unroll
            for (int j = 0; j < 8; ++j)
                C[(size_t)(mBase + j) * NN + t * 16 + nlane] = acc[t][j];
    }
}

// ---- layer-1 attention logits: one thread per (node, head), 32-dot ----
__global__ void logits1(const float* __restrict__ h1, const float* __restrict__ att_s,
                        const float* __restrict__ att_d, float* __restrict__ as_,
                        float* __restrict__ ad_) {
    int t = blockIdx.x * blockDim.x + threadIdx.x;
    if (t >= N_NODES * HEADS) return;
    int n = t >> 2, h = t & 3;
    const float* hp = h1 + (size_t)n * H1DIM + h * HID;
    float ss = 0.f, dd = 0.f;
#pragma unroll
    for (int c = 0; c < HID; ++c) {
        float v = hp[c];
        ss += v * att_s[h * HID + c];
        dd += v * att_d[h * HID + c];
    }
    as_[t] = ss; ad_[t] = dd;
}

__global__ void edge_max1(const int* __restrict__ ei, const float* __restrict__ as_,
                          const float* __restrict__ ad_, float* __restrict__ m) {
    int e = blockIdx.x * blockDim.x + threadIdx.x;
    if (e >= E_TOT) return;
    int s, d; edge_sd(ei, e, s, d);
    float4 av = ((const float4*)as_)[s];
    float4 dv = ((const float4*)ad_)[d];
    atomicMaxF(&m[d * HEADS + 0], lrelu(av.x + dv.x));
    atomicMaxF(&m[d * HEADS + 1], lrelu(av.y + dv.y));
    atomicMaxF(&m[d * HEADS + 2], lrelu(av.z + dv.z));
    atomicMaxF(&m[d * HEADS + 3], lrelu(av.w + dv.w));
}

__global__ void edge_sum1(const int* __restrict__ ei, const float* __restrict__ as_,
                          const float* __restrict__ ad_, const float* __restrict__ m,
                          float* __restrict__ ssum) {
    int e = blockIdx.x * blockDim.x + threadIdx.x;
    if (e >= E_TOT) return;
    int s, d; edge_sd(ei, e, s, d);
    float4 av = ((const float4*)as_)[s];
    float4 dv = ((const float4*)ad_)[d];
    float4 mv = ((const float4*)m)[d];
    atomicAdd(&ssum[d * HEADS + 0], __expf(lrelu(av.x + dv.x) - mv.x));
    atomicAdd(&ssum[d * HEADS + 1], __expf(lrelu(av.y + dv.y) - mv.y));
    atomicAdd(&ssum[d * HEADS + 2], __expf(lrelu(av.z + dv.z) - mv.z));
    atomicAdd(&ssum[d * HEADS + 3], __expf(lrelu(av.w + dv.w) - mv.w));
}

// ---- weighted scatter-add: 32 lanes per edge, float4 per lane ----
__global__ void aggregate1(const int* __restrict__ ei, const float* __restrict__ h1,
                           const float* __restrict__ as_, const float* __restrict__ ad_,
                           const float* __restrict__ m, const float* __restrict__ ssum,
                           float* __restrict__ agg) {
    long long t = (long long)blockIdx.x * blockDim.x + threadIdx.x;
    if (t >= (long long)E_TOT * 32) return;
    int e = (int)(t >> 5);
    int lane = (int)(t & 31);
    int h = lane >> 3;                        // 8 lanes per head (32 ch / 4)
    int s, d; edge_sd(ei, e, s, d);
    float l = lrelu(as_[s * HEADS + h] + ad_[d * HEADS + h]);
    float w = __expf(l - m[d * HEADS + h]) / (ssum[d * HEADS + h] + 1e-16f);
    float4 v = ((const float4*)h1)[(size_t)s * 32 + lane];
    float* out = agg + (size_t)d * H1DIM + lane * 4;
    atomicAdd(out + 0, w * v.x);
    atomicAdd(out + 1, w * v.y);
    atomicAdd(out + 2, w * v.z);
    atomicAdd(out + 3, w * v.w);
}

__global__ void finish1(float* __restrict__ agg, const float* __restrict__ b1) {
    size_t i = (size_t)blockIdx.x * blockDim.x + threadIdx.x;
    if (i >= (size_t)N_NODES * H1DIM) return;
    float v = agg[i] + b1[i & (H1DIM - 1)];
    agg[i] = v > 0.f ? v : 0.f;
}

// ---- layer 2 (1 head, 16 channels) ----
__global__ void logits2(const float* __restrict__ g2, const float* __restrict__ att_s,
                        const float* __restrict__ att_d, float* __restrict__ as_,
                        float* __restrict__ ad_) {
    int n = blockIdx.x * blockDim.x + threadIdx.x;
    if (n >= N_NODES) return;
    const float* gp = g2 + (size_t)n * C2;
    float ss = 0.f, dd = 0.f;
#pragma unroll
    for (int c = 0; c < C2; ++c) { float v = gp[c]; ss += v * att_s[c]; dd += v * att_d[c]; }
    as_[n] = ss; ad_[n] = dd;
}

__global__ void edge_max2(const int* __restrict__ ei, const float* __restrict__ as_,
                          const float* __restrict__ ad_, float* __restrict__ m) {
    int e = blockIdx.x * blockDim.x + threadIdx.x;
    if (e >= E_TOT) return;
    int s, d; edge_sd(ei, e, s, d);
    atomicMaxF(&m[d], lrelu(as_[s] + ad_[d]));
}

__global__ void edge_sum2(const int* __restrict__ ei, const float* __restrict__ as_,
                          const float* __restrict__ ad_, const float* __restrict__ m,
                          float* __restrict__ ssum) {
    int e = blockIdx.x * blockDim.x + threadIdx.x;
    if (e >= E_TOT) return;
    int s, d; edge_sd(ei, e, s, d);
    atomicAdd(&ssum[d], __expf(lrelu(as_[s] + ad_[d]) - m[d]));
}

__global__ void aggregate2(const int* __restrict__ ei, const float* __restrict__ g2,
                           const float* __restrict__ as_, const float* __restrict__ ad_,
                           const float* __restrict__ m, const float* __restrict__ ssum,
                           float* __restrict__ agg) {
    long long t = (long long)blockIdx.x * blockDim.x + threadIdx.x;
    if (t >= (long long)E_TOT * 4) return;
    int e = (int)(t >> 2);
    int q = (int)(t & 3);
    int s, d; edge_sd(ei, e, s, d);
    float w = __expf(lrelu(as_[s] + ad_[d]) - m[d]) / (ssum[d] + 1e-16f);
    float4 v = ((const float4*)g2)[(size_t)s * 4 + q];
    float* out = agg + (size_t)d * C2 + q * 4;
    atomicAdd(out + 0, w * v.x);
    atomicAdd(out + 1, w * v.y);
    atomicAdd(out + 2, w * v.z);
    atomicAdd(out + 3, w * v.w);
}

__global__ void finish2_pool(const float* __restrict__ agg2, const float* __restrict__ b2,
                             const int* __restrict__ nodeIDs, float* __restrict__ pool,
                             float* __restrict__ cnt) {
    size_t t = (size_t)blockIdx.x * blockDim.x + threadIdx.x;
    if (t >= (size_t)N_NODES * C2) return;
    int n = (int)(t >> 4), c = (int)(t & 15);
    float v = agg2[t] + b2[c];
    v = v > 0.f ? v : 0.f;
    int g = nodeIDs[n];
    atomicAdd(&pool[g * C2 + c], v);
    if (c == 0) atomicAdd(&cnt[g], 1.0f);
}

__global__ void final_gemm(const float* __restrict__ pool, const float* __restrict__ cnt,
                           const float* __restrict__ Wf, const float* __restrict__ bf,
                           float* __restrict__ out) {
    int t = blockIdx.x * blockDim.x + threadIdx.x;
    if (t >= N_GRAPHS * OUT_CH) return;
    int g = t / OUT_CH, o = t % OUT_CH;
    float inv = 1.0f / fmaxf(cnt[g], 1.0f);
    float acc = bf[o];
#pragma unroll
    for (int c = 0; c < C2; ++c)
        acc += pool[g * C2 + c] * inv * Wf[c * OUT_CH + o];
    out[t] = acc;
}

extern "C" void kernel_launch(void* const* d_in, const int* in_sizes, int n_in,
                              void* d_out, int out_size, void* d_ws, size_t ws_size,
                              hipStream_t stream) {
    const float* x        = (const float*)d_in[0];
    const int*   ei       = (const int*)  d_in[1];
    // d_in[2] = edge_attr (unused by reference GATConv)
    const int*   nodeIDs  = (const int*)  d_in[3];
    const float* W1       = (const float*)d_in[4];
    const float* att_src1 = (const float*)d_in[5];
    const float* att_dst1 = (const float*)d_in[6];
    const float* b1       = (const float*)d_in[7];
    const float* W2       = (const float*)d_in[8];
    const float* att_src2 = (const float*)d_in[9];
    const float* att_dst2 = (const float*)d_in[10];
    const float* b2       = (const float*)d_in[11];
    const float* Wf       = (const float*)d_in[12];
    const float* bf       = (const float*)d_in[13];
    float* out = (float*)d_out;

    float* ws   = (float*)d_ws;
    float* h1   = ws + OFF_H1;
    float* agg1 = ws + OFF_AGG1;
    float* as1  = ws + OFF_AS1;
    float* ad1  = ws + OFF_AD1;
    float* m1   = ws + OFF_M1;
    float* s1   = ws + OFF_S1;
    float* g2   = ws + OFF_G2;
    float* agg2 = ws + OFF_AGG2;
    float* as2  = ws + OFF_AS2;
    float* ad2  = ws + OFF_AD2;
    float* m2   = ws + OFF_M2;
    float* s2   = ws + OFF_S2;
    float* pool = ws + OFF_POOL;
    float* cnt  = ws + OFF_CNT;

    const int B = 256;
    const int MT = N_NODES / 16;   // 3125 M-tiles (exact)
    init_ws<<<2048, B, 0, stream>>>(ws);

    // layer 1: h1 = x @ W1  (wave = 16x128 row-block, W1 staged in LDS)
    wmma_gemm_lds<8, IN_CH><<<(MT + 7) / 8, B, 0, stream>>>(x, W1, h1, MT);
    logits1<<<(N_NODES * HEADS + B - 1) / B, B, 0, stream>>>(h1, att_src1, att_dst1, as1, ad1);
    edge_max1<<<(E_TOT + B - 1) / B, B, 0, stream>>>(ei, as1, ad1, m1);
    edge_sum1<<<(E_TOT + B - 1) / B, B, 0, stream>>>(ei, as1, ad1, m1, s1);
    {
        long long tot = (long long)E_TOT * 32;
        aggregate1<<<(unsigned)((tot + B - 1) / B), B, 0, stream>>>(ei, h1, as1, ad1, m1, s1, agg1);
    }
    finish1<<<(unsigned)(((size_t)N_NODES * H1DIM + B - 1) / B), B, 0, stream>>>(agg1, b1);

    // layer 2: g2 = relu1 @ W2 (N=16 -> single N-tile per wave)
    wmma_gemm_lds<1, H1DIM><<<(MT + 7) / 8, B, 0, stream>>>(agg1, W2, g2, MT);
    logits2<<<(N_NODES + B - 1) / B, B, 0, stream>>>(g2, att_src2, att_dst2, as2, ad2);
    edge_max2<<<(E_TOT + B - 1) / B, B, 0, stream>>>(ei, as2, ad2, m2);
    edge_sum2<<<(E_TOT + B - 1) / B, B, 0, stream>>>(ei, as2, ad2, m2, s2);
    {
        long long tot = (long long)E_TOT * 4;
        aggregate2<<<(unsigned)((tot + B - 1) / B), B, 0, stream>>>(ei, g2, as2, ad2, m2, s2, agg2);
    }
    finish2_pool<<<(unsigned)(((size_t)N_NODES * C2 + B - 1) / B), B, 0, stream>>>(agg2, b2, nodeIDs, pool, cnt);
    final_gemm<<<(N_GRAPHS * OUT_CH + B - 1) / B, B, 0, stream>>>(pool, cnt, Wf, bf, out);
}